// AdaptivePruner_36558761624178
// MI455X (gfx1250) — compile-verified
//
#include <hip/hip_runtime.h>

// ---------------------------------------------------------------------------
// AdaptivePruner for MI455X (gfx1250, wave32)
//
// out = (final_x[256,102,768] f32, mask[256,102] f32) concatenated flat.
//
// Both DWT levels are expressed as dense banded matrices W[2][112][196]
// (fp32, built analytically on device, L2-resident); then
//   final_patches(b) = W[level(b)] @ patch(b)   via V_WMMA_F32_16X16X4_F32
// (exact fp32 math on the matrix pipe).
//
// Roofline: ~235 MB mandatory HBM traffic => ~10 us at 23.3 TB/s; FLOPs are
// negligible. The GEMM is register-blocked over M-tiles (MT=3) so each
// B-fragment load feeds 3 WMMAs, cutting activation re-reads from 7x to 3x
// (~450 MB of L2 traffic, below the HBM floor).
// ---------------------------------------------------------------------------

typedef float v2f __attribute__((ext_vector_type(2)));
typedef float v8f __attribute__((ext_vector_type(8)));

#define B_BATCH 256
#define N_TOK   197
#define D_DIM   768
#define N_PATCH 196
#define LEN1    101      // rows after 1 DWT level
#define LEN2    54       // rows after 2 DWT levels
#define M_PAD   112      // 7*16
#define K_DIM   196
#define OUT_TOK 102      // 1 cls + 101 patch rows
#define SOGLIA  0.333f

__constant__ float DB4_H_c[8] = {
    0.23037781330885523f,  0.7148465705525415f,  0.6308807679295904f,
   -0.02798376941698385f, -0.18703481171888114f, 0.030841381835986965f,
    0.032883011666982945f, -0.010597401784997278f
};

// ---------------------------------------------------------------------------
// Build W[0] = A1 (one-level DWT matrix, 101x196 banded, zero rows 101..111)
//       W[1] = W2*A1 (two-level composition, 54x196, zero rows 54..111)
// y1[n] = sum_l H[l] * xpad[2n+l], xpad has 6 zeros each side  => l = c-2n+6
// y2[n] = sum_l H[l] * y1ext[2n+l], y1ext: y1 + 1 zero at end + 6 each side
// ---------------------------------------------------------------------------
__global__ void build_w_kernel(float* __restrict__ Wmat) {
    int idx = blockIdx.x * blockDim.x + threadIdx.x;
    const int total = 2 * M_PAD * K_DIM;
    if (idx >= total) return;
    int lvl = idx / (M_PAD * K_DIM);
    int rem = idx % (M_PAD * K_DIM);
    int n = rem / K_DIM;
    int c = rem % K_DIM;
    float v = 0.f;
    if (lvl == 0) {
        int l = c - 2 * n + 6;
        if (n < LEN1 && l >= 0 && l < 8) v = DB4_H_c[l];
    } else {
        if (n < LEN2) {
            #pragma unroll
            for (int l = 0; l < 8; ++l) {
                int m = 2 * n + l - 6;          // index into y1 (0..100 valid)
                if (m >= 0 && m < LEN1) {
                    int l1 = c - 2 * m + 6;
                    if (l1 >= 0 && l1 < 8) v += DB4_H_c[l] * DB4_H_c[l1];
                }
            }
        }
    }
    Wmat[idx] = v;
}

// ---------------------------------------------------------------------------
// Per-batch: bitonic-sort the 196 attention values in LDS, compute gini,
// pick the level, write the mask row, and copy the cls token row.
// One 256-thread block (8 waves) per batch.
// ---------------------------------------------------------------------------
__global__ void gini_mask_cls_kernel(const float* __restrict__ attn,
                                     const float* __restrict__ x,
                                     float* __restrict__ out,
                                     int* __restrict__ lvlbuf) {
    const int b   = blockIdx.x;
    const int tid = threadIdx.x;

    __shared__ float s[256];
    __shared__ float red[256];
    __shared__ int   slvl;

    s[tid] = (tid < N_PATCH) ? attn[(size_t)b * N_PATCH + tid] : 3.0e38f;
    __syncthreads();

    // Bitonic sort, ascending; padding (+huge) lands at positions 196..255.
    for (int k = 2; k <= 256; k <<= 1) {
        for (int j = k >> 1; j > 0; j >>= 1) {
            int ixj = tid ^ j;
            if (ixj > tid) {
                float a = s[tid], c = s[ixj];
                bool up = ((tid & k) == 0);
                if ((a > c) == up) { s[tid] = c; s[ixj] = a; }
            }
            __syncthreads();
        }
    }

    float v  = (tid < N_PATCH) ? s[tid] : 0.f;
    float wv = (tid < N_PATCH) ? (float)(tid + 1) * s[tid] : 0.f;

    red[tid] = v;
    __syncthreads();
    for (int off = 128; off > 0; off >>= 1) {
        if (tid < off) red[tid] += red[tid + off];
        __syncthreads();
    }
    float sum = red[0];
    __syncthreads();

    red[tid] = wv;
    __syncthreads();
    for (int off = 128; off > 0; off >>= 1) {
        if (tid < off) red[tid] += red[tid + off];
        __syncthreads();
    }
    float wsum = red[0];

    if (tid == 0) {
        float gini = 2.f * wsum / ((float)N_PATCH * sum + 1e-8f)
                   - (float)(N_PATCH + 1) / (float)N_PATCH;
        slvl = (gini > SOGLIA) ? 1 : 0;
        lvlbuf[b] = slvl;
    }
    __syncthreads();
    const int lvl    = slvl;
    const int outlen = lvl ? LEN2 : LEN1;

    // mask row: [1, (i < outlen) for i in 0..100]
    float* mask = out + (size_t)B_BATCH * OUT_TOK * D_DIM + (size_t)b * OUT_TOK;
    if (tid == 0)   mask[0] = 1.f;
    if (tid < LEN1) mask[1 + tid] = (tid < outlen) ? 1.f : 0.f;

    // cls token copy: out[b,0,:] = x[b,0,:]
    const float* xc = x   + (size_t)b * N_TOK   * D_DIM;
    float*       oc = out + (size_t)b * OUT_TOK * D_DIM;
    for (int d = tid; d < D_DIM; d += 256) oc[d] = xc[d];
}

// ---------------------------------------------------------------------------
// final_patches(b) = W[level(b)] (112x196) @ patch(b) (196x768)
//
// Register-blocked GEMM: template <MT, MTBASE>; each block handles M-tiles
// MTBASE..MTBASE+MT-1 for one batch; 8 waves, each owning a (MT*16 x 96)
// output chunk (MT*6 v8f accumulators). One B-fragment load feeds MT WMMAs.
//
//   A 16x4 frag (2 VGPR): lanes 0-15 -> K0/K1, lanes 16-31 -> K2/K3, M=lane&15
//   B 4x16 frag (2 VGPR): same K split, N = lane&15
//   D 16x16  (8 VGPR):    VGPR r -> M = r + 8*(lane>>4), N = lane&15
// ---------------------------------------------------------------------------
template <int MT, int MTBASE>
__global__ void __launch_bounds__(256)
dwt_gemm_kernel(const float* __restrict__ x,
                const float* __restrict__ Wmat,
                const int*   __restrict__ lvlbuf,
                float* __restrict__ out) {
    const int b    = blockIdx.x;        // 0..255
    const int wave = threadIdx.x >> 5;  // 0..7
    const int lane = threadIdx.x & 31;
    const int lm   = lane & 15;
    const int kh   = lane >> 4;         // 0: K rows {0,1}; 1: K rows {2,3}

    const int lvl = lvlbuf[b];
    const float* Wbase = Wmat + (size_t)lvl * (M_PAD * K_DIM);
    const float* Wm[MT];
    #pragma unroll
    for (int mtt = 0; mtt < MT; ++mtt)
        Wm[mtt] = Wbase + (size_t)((MTBASE + mtt) * 16 + lm) * K_DIM;

    const float* Xb = x + (size_t)b * N_TOK * D_DIM + D_DIM;  // skip cls row

    v8f acc[MT][6];
    #pragma unroll
    for (int mtt = 0; mtt < MT; ++mtt)
        #pragma unroll
        for (int t = 0; t < 6; ++t)
            #pragma unroll
            for (int i = 0; i < 8; ++i) acc[mtt][t][i] = 0.f;

    const int n0 = wave * 96;           // this wave's 96-column block

    for (int k = 0; k < 49; ++k) {      // K = 196 = 49 * 4
        const int kb = k * 4 + kh * 2;

        v2f a[MT];                      // A fragments (weight matrix, L2-hot)
        #pragma unroll
        for (int mtt = 0; mtt < MT; ++mtt) {
            a[mtt].x = Wm[mtt][kb + 0];
            a[mtt].y = Wm[mtt][kb + 1];
        }

        const float* xr0 = Xb + (size_t)(kb + 0) * D_DIM;
        const float* xr1 = xr0 + D_DIM;

        // prefetch activation panel two K-chunks ahead
        if (k + 2 < 49)
            __builtin_prefetch(xr0 + (size_t)8 * D_DIM + n0 + lm, 0, 1);

        v2f bf[6];                      // B fragments, shared by all M-tiles
        #pragma unroll
        for (int t = 0; t < 6; ++t) {
            const int n = n0 + t * 16 + lm;
            bf[t].x = xr0[n];
            bf[t].y = xr1[n];
        }

        #pragma unroll
        for (int mtt = 0; mtt < MT; ++mtt)
            #pragma unroll
            for (int t = 0; t < 6; ++t)
                acc[mtt][t] = __builtin_amdgcn_wmma_f32_16x16x4_f32(
                    false, a[mtt], false, bf[t], (short)0, acc[mtt][t],
                    false, false);
    }

    float* Ob = out + (size_t)b * OUT_TOK * D_DIM;
    #pragma unroll
    for (int mtt = 0; mtt < MT; ++mtt) {
        #pragma unroll
        for (int t = 0; t < 6; ++t) {
            const int n = n0 + t * 16 + lm;
            #pragma unroll
            for (int r = 0; r < 8; ++r) {
                const int m = (MTBASE + mtt) * 16 + r + kh * 8;
                // guard folds away for tiles fully inside [0,101)
                if ((MTBASE + mtt) * 16 + 15 < LEN1 || m < LEN1)
                    Ob[(size_t)(1 + m) * D_DIM + n] = acc[mtt][t][r];
            }
        }
    }
}

// ---------------------------------------------------------------------------
extern "C" void kernel_launch(void* const* d_in, const int* in_sizes, int n_in,
                              void* d_out, int out_size, void* d_ws, size_t ws_size,
                              hipStream_t stream) {
    const float* x    = (const float*)d_in[0];   // (256,197,768) f32
    const float* attn = (const float*)d_in[1];   // (256,196)     f32
    float*       out  = (float*)d_out;

    float* Wmat   = (float*)d_ws;                              // 2*112*196 f32
    int*   lvlbuf = (int*)((char*)d_ws +
                           (size_t)2 * M_PAD * K_DIM * sizeof(float)); // 256 ints

    {
        const int total = 2 * M_PAD * K_DIM;
        build_w_kernel<<<(total + 255) / 256, 256, 0, stream>>>(Wmat);
    }

    gini_mask_cls_kernel<<<B_BATCH, 256, 0, stream>>>(attn, x, out, lvlbuf);

    // M-tiles 0..2, 3..5 (unguarded stores), tile 6 (guarded rows 96..100)
    dwt_gemm_kernel<3, 0><<<B_BATCH, 256, 0, stream>>>(x, Wmat, lvlbuf, out);
    dwt_gemm_kernel<3, 3><<<B_BATCH, 256, 0, stream>>>(x, Wmat, lvlbuf, out);
    dwt_gemm_kernel<1, 6><<<B_BATCH, 256, 0, stream>>>(x, Wmat, lvlbuf, out);
}